// Author_11982958756126
// MI455X (gfx1250) — compile-verified
//
#include <hip/hip_runtime.h>

// Embedding gather: out[t, :] = W[:, idx[t]] + b      (B*S tokens, E=128, V=1e6)
// Pure memory-bound scatter-gather (~0.008 FLOP/B): every gathered element
// W[e*V + idx] sits on its own HBM line (4MB stride), ~1GB of line traffic,
// floor ~46us at 23.3 TB/s. Optimization = in-flight-line count + cache
// pollution control, not compute. No WMMA (would be a K=1e6 one-hot GEMM,
// ~16 TFLOP of wasted matrix work for 8M useful adds).
//
// CDNA5 paths used: th:TH_NT temporal hints on the zero-reuse gather and the
// streamed 32MB output, global_prefetch_b8 on upcoming index lines, wave32
// layout with readfirstlane-scalarized token indices so gathers use the
// SGPR-base + 32-bit VGPR-offset (GVS) form.

#define EMB             128
#define VOCAB           1000000
#define TOK_PER_WAVE    8
#define WAVES_PER_BLOCK 8
#define BLOCK_THREADS   (WAVES_PER_BLOCK * 32)

typedef float v4f __attribute__((ext_vector_type(4)));
typedef int   v4i __attribute__((ext_vector_type(4)));

__global__ __launch_bounds__(BLOCK_THREADS) void
Author_11982958756126_gather_emb(const int* __restrict__ idx,
                                 const float* __restrict__ W,
                                 const float* __restrict__ b,
                                 float* __restrict__ out,
                                 int n_tokens)
{
    const int lane = threadIdx.x & 31;
    const int waveInBlock = threadIdx.x >> 5;
    const int wave = blockIdx.x * WAVES_PER_BLOCK + waveInBlock;
    const int tok0 = wave * TOK_PER_WAVE;
    if (tok0 >= n_tokens) return;

    // Lane l owns emb dims 4l..4l+3: one b128 NT store per lane per token,
    // a fully-coalesced 512B burst per wave.
    const int e0 = lane * 4;

    // Per-lane gather row offsets (elements). Max offset 127e6 + 1e6 < 2^31,
    // so all address math stays 32-bit -> saddr(W) + u32 voffset loads.
    unsigned row_off[4];
#pragma unroll
    for (int j = 0; j < 4; ++j)
        row_off[j] = (unsigned)(e0 + j) * (unsigned)VOCAB;

    // Bias: 512B, re-read by every wave -> default RT hint, stays in WGP$.
    const v4f bias = *reinterpret_cast<const v4f*>(b + e0);

    // Prefetch the next token-group's index line.
    if (tok0 + TOK_PER_WAVE < n_tokens)
        __builtin_prefetch(idx + tok0 + TOK_PER_WAVE, 0, 1);

    if (tok0 + TOK_PER_WAVE <= n_tokens) {
        // ---- Fast path: branch-free, always taken for the 65536-token shape.
        // Wave's 8 indices: two b128 loads of the same 32B line (coalesces),
        // then scalarize with readfirstlane.
        const v4i* ip = reinterpret_cast<const v4i*>(idx + tok0);
        const v4i q0 = ip[0];
        const v4i q1 = ip[1];
        int tix[TOK_PER_WAVE];
        tix[0] = __builtin_amdgcn_readfirstlane(q0.x);
        tix[1] = __builtin_amdgcn_readfirstlane(q0.y);
        tix[2] = __builtin_amdgcn_readfirstlane(q0.z);
        tix[3] = __builtin_amdgcn_readfirstlane(q0.w);
        tix[4] = __builtin_amdgcn_readfirstlane(q1.x);
        tix[5] = __builtin_amdgcn_readfirstlane(q1.y);
        tix[6] = __builtin_amdgcn_readfirstlane(q1.z);
        tix[7] = __builtin_amdgcn_readfirstlane(q1.w);

        // Issue ALL 32 gathered loads before any use: 32 lines in flight per
        // lane. NT: random columns of a 512MB table, ~3% reuse -> don't evict
        // the hot bias/index lines from WGP$/L2.
        float v[TOK_PER_WAVE][4];
#pragma unroll
        for (int k = 0; k < TOK_PER_WAVE; ++k) {
#pragma unroll
            for (int j = 0; j < 4; ++j)
                v[k][j] = __builtin_nontemporal_load(
                    W + (row_off[j] + (unsigned)tix[k]));
        }

        // Bias-add and stream out (written once, never re-read on device).
#pragma unroll
        for (int k = 0; k < TOK_PER_WAVE; ++k) {
            v4f r;
            r.x = v[k][0] + bias.x;
            r.y = v[k][1] + bias.y;
            r.z = v[k][2] + bias.z;
            r.w = v[k][3] + bias.w;
            __builtin_nontemporal_store(
                r, reinterpret_cast<v4f*>(out + (size_t)(tok0 + k) * EMB + e0));
        }
    } else {
        // ---- Tail (never taken for 32x2048 tokens; kept for generality).
        for (int t = tok0; t < n_tokens; ++t) {
            const int ti = __builtin_amdgcn_readfirstlane(idx[t]);
            v4f r;
            r.x = __builtin_nontemporal_load(W + (row_off[0] + (unsigned)ti)) + bias.x;
            r.y = __builtin_nontemporal_load(W + (row_off[1] + (unsigned)ti)) + bias.y;
            r.z = __builtin_nontemporal_load(W + (row_off[2] + (unsigned)ti)) + bias.z;
            r.w = __builtin_nontemporal_load(W + (row_off[3] + (unsigned)ti)) + bias.w;
            __builtin_nontemporal_store(
                r, reinterpret_cast<v4f*>(out + (size_t)t * EMB + e0));
        }
    }
}

extern "C" void kernel_launch(void* const* d_in, const int* in_sizes, int n_in,
                              void* d_out, int out_size, void* d_ws, size_t ws_size,
                              hipStream_t stream) {
    const int*   idx = (const int*)  d_in[0];   // bow_vec, [B*S]
    const float* W   = (const float*)d_in[1];   // [E, V]
    const float* b   = (const float*)d_in[2];   // [E]
    float*       out = (float*)      d_out;     // [B*S, E]

    const int n_tokens = in_sizes[0];
    const int tokens_per_block = TOK_PER_WAVE * WAVES_PER_BLOCK;      // 64
    const int blocks = (n_tokens + tokens_per_block - 1) / tokens_per_block;

    Author_11982958756126_gather_emb<<<blocks, BLOCK_THREADS, 0, stream>>>(
        idx, W, b, out, n_tokens);
}